// GCNOne_4011499454554
// MI455X (gfx1250) — compile-verified
//
#include <hip/hip_runtime.h>
#include <hip/hip_bf16.h>
#include <math.h>

// ---------------------------------------------------------------------------
// GCN (4x GCNConv + tanh) for MI455X / gfx1250, wave32.
//  - GEMMs: V_WMMA_F32_16X16X4_F32, k-pair-interleaved weight tile in LDS so
//    every B operand is one conflict-free ds_load_b64 (no repack movs);
//    2 row-tiles per wave => each B load feeds 2 WMMAs.
//  - GEMM epilogue fused with the GCN self-loop term: writes h AND
//    acc = h*dinv^2 + b in one pass (saves a full N*128 read+write per layer).
//  - Edge aggregation: hardware fp32 atomics (GLOBAL_ATOMIC_ADD_F32) on an
//    L2-resident 51 MB feature matrix (192 MB L2), plus global_prefetch_b8
//    on the gather rows a few hundred edges ahead.
// ---------------------------------------------------------------------------

#define HDIM 128
#define WS2  144   // stride (in float2) of paired W tile: 2*144 % 64 == 32 -> halves disjoint
#define A_LD 138   // A tile stride in floats: even (8B-aligned b64), conflict-free

typedef __attribute__((ext_vector_type(2))) float v2f;
typedef __attribute__((ext_vector_type(8))) float v8f;

// ---------------------------------------------------------------------------
// h[N,128] = A[N,K] * W[K,128];  acc[N,128] = h * dinv[row]^2 + bias[col]
// 256 threads = 8 waves; block covers 256 rows; each wave: 32 rows (2 WMMA
// row-tiles) x 128 cols (8 col-tiles).
// ---------------------------------------------------------------------------
__global__ __launch_bounds__(256)
void gcn_gemm_wmma_f32(const float* __restrict__ A,
                       const float* __restrict__ W,
                       const float* __restrict__ bias,
                       const float* __restrict__ dinv,
                       float* __restrict__ C,
                       float* __restrict__ ACC,
                       int N, int K, int Kpad) {
    extern __shared__ float lds[];
    v2f*   Wp = (v2f*)lds;                           // [Kpad/2][WS2] pairs (k, k+1)
    float* As = lds + (size_t)(Kpad / 2) * WS2 * 2;  // 8 waves x [32][A_LD]

    const int tid  = threadIdx.x;
    const int wave = tid >> 5;
    const int lane = tid & 31;
    const int l15  = lane & 15;
    const int rowBase = blockIdx.x * 256 + wave * 32;

    // Stage W into LDS, k-pair interleaved: Wp[kk][n] = (W[2kk][n], W[2kk+1][n]).
    for (int idx = tid; idx < (Kpad / 2) * HDIM; idx += 256) {
        int kk = idx >> 7;
        int n  = idx & (HDIM - 1);
        int k0 = kk * 2;
        v2f w;
        w.x = (k0     < K) ? W[(size_t)k0       * HDIM + n] : 0.0f;
        w.y = (k0 + 1 < K) ? W[(size_t)(k0 + 1) * HDIM + n] : 0.0f;
        Wp[kk * WS2 + n] = w;
    }

    // Stage this wave's 32-row A strip into LDS, zero-padded.
    float* Aw = As + (size_t)wave * 32 * A_LD;
    for (int r = 0; r < 32; ++r) {
        int row = rowBase + r;
        for (int k = lane; k < Kpad; k += 32) {
            Aw[r * A_LD + k] = (row < N && k < K) ? A[(size_t)row * K + k] : 0.0f;
        }
    }
    __syncthreads();

    v8f acc[2][8];
#pragma unroll
    for (int t = 0; t < 2; ++t)
#pragma unroll
        for (int j = 0; j < 8; ++j)
            acc[t][j] = (v8f){0.f,0.f,0.f,0.f,0.f,0.f,0.f,0.f};

    // ISA 16x4 f32 A layout: lanes 0-15 -> K = {k0, k0+1}, lanes 16-31 -> {k0+2, k0+3}
    const int koff = (lane >= 16) ? 2 : 0;
    const float* Aw0 = Aw + l15 * A_LD;          // tile 0: rows rowBase + l15
    const float* Aw1 = Aw + (l15 + 16) * A_LD;   // tile 1: rows rowBase + 16 + l15

    for (int k0 = 0; k0 < Kpad; k0 += 4) {
        v2f a0 = *(const v2f*)(Aw0 + k0 + koff);
        v2f a1 = *(const v2f*)(Aw1 + k0 + koff);
        const v2f* Bk = Wp + (size_t)((k0 + koff) >> 1) * WS2 + l15;
#pragma unroll
        for (int j = 0; j < 8; ++j) {
            v2f b = Bk[j * 16];
            acc[0][j] = __builtin_amdgcn_wmma_f32_16x16x4_f32(
                false, a0, false, b, (short)0, acc[0][j], false, false);
            acc[1][j] = __builtin_amdgcn_wmma_f32_16x16x4_f32(
                false, a1, false, b, (short)0, acc[1][j], false, false);
        }
    }

    // Epilogue: write h and the GCN self-loop seeded accumulator.
    // C/D layout: VGPR i holds row M=i (lanes 0-15) / M=i+8 (lanes 16-31), col N=lane&15.
    float bcol[8];
#pragma unroll
    for (int j = 0; j < 8; ++j) bcol[j] = bias[j * 16 + l15];

    const int rOff = (lane >= 16) ? 8 : 0;
#pragma unroll
    for (int t = 0; t < 2; ++t) {
#pragma unroll
        for (int i = 0; i < 8; ++i) {
            int row = rowBase + t * 16 + i + rOff;
            if (row < N) {
                float di  = dinv[row];
                float di2 = di * di;
                size_t base = (size_t)row * HDIM + l15;
#pragma unroll
                for (int j = 0; j < 8; ++j) {
                    float h = acc[t][j][i];
                    C[base + j * 16]   = h;
                    ACC[base + j * 16] = h * di2 + bcol[j];
                }
            }
        }
    }
}

// ---------------------------------------------------------------------------
// Degree / dinv helpers
// ---------------------------------------------------------------------------
__global__ void zero_f32(float* __restrict__ p, int n) {
    int i = blockIdx.x * blockDim.x + threadIdx.x;
    if (i < n) p[i] = 0.0f;
}

__global__ void deg_count(const int* __restrict__ dst, float* __restrict__ deg, int E) {
    int e = blockIdx.x * blockDim.x + threadIdx.x;
    if (e < E) unsafeAtomicAdd(&deg[dst[e]], 1.0f);
}

__global__ void finalize_dinv(float* __restrict__ d, int n) {
    int i = blockIdx.x * blockDim.x + threadIdx.x;
    if (i < n) d[i] = rsqrtf(d[i] + 1.0f);
}

// ---------------------------------------------------------------------------
// One wave (32 lanes) per edge; lane handles 4 features via float4 gather
// and 4 hardware fp32 atomic adds into the destination row. Prefetch the
// gather row for an edge ~256 ahead (L2 -> WGP).
// ---------------------------------------------------------------------------
__global__ __launch_bounds__(256)
void edge_scatter(const int* __restrict__ src, const int* __restrict__ dst,
                  const float* __restrict__ dinv, const float* __restrict__ g,
                  float* __restrict__ acc, int E) {
    int e = blockIdx.x * 8 + (threadIdx.x >> 5);
    if (e >= E) return;
    int lane = threadIdx.x & 31;

    int ep = e + 256;
    if (ep < E) {
        int sp = src[ep];
        __builtin_prefetch(g + (size_t)sp * HDIM + lane * 4, 0, 1);
    }

    int s = src[e];
    int d = dst[e];
    float coef = dinv[s] * dinv[d];
    const float4* gs = (const float4*)(g + (size_t)s * HDIM);
    float4 v = gs[lane];
    float* ad = acc + (size_t)d * HDIM + lane * 4;
    unsafeAtomicAdd(ad + 0, coef * v.x);
    unsafeAtomicAdd(ad + 1, coef * v.y);
    unsafeAtomicAdd(ad + 2, coef * v.z);
    unsafeAtomicAdd(ad + 3, coef * v.w);
}

__global__ void tanh_act(const float* __restrict__ in, float* __restrict__ out, int n) {
    int i = blockIdx.x * blockDim.x + threadIdx.x;
    if (i < n) out[i] = tanhf(in[i]);
}

// ---------------------------------------------------------------------------
extern "C" void kernel_launch(void* const* d_in, const int* in_sizes, int n_in,
                              void* d_out, int out_size, void* d_ws, size_t ws_size,
                              hipStream_t stream) {
    const float* x  = (const float*)d_in[0];
    const int*   ei = (const int*)d_in[1];
    const float* Wl[4] = {(const float*)d_in[2], (const float*)d_in[4],
                          (const float*)d_in[6], (const float*)d_in[8]};
    const float* bl[4] = {(const float*)d_in[3], (const float*)d_in[5],
                          (const float*)d_in[7], (const float*)d_in[9]};

    const int NFEAT = in_sizes[2] / HDIM;       // 133
    const int N     = in_sizes[0] / NFEAT;      // 100000
    const int E     = in_sizes[1] / 2;          // 1600000
    const int* srcp = ei;
    const int* dstp = ei + E;

    // Workspace layout: dinv[N] | r0[N*H] (h) | r1[N*H] (accumulator) | r2[N*H] (tanh out)
    float* wsf  = (float*)d_ws;
    float* dinv = wsf;
    float* r0   = wsf + N;
    float* r1   = r0 + (size_t)N * HDIM;
    float* r2   = r1 + (size_t)N * HDIM;

    const int THR = 256;

    // dinv = rsqrt(deg + 1)
    zero_f32<<<(N + THR - 1) / THR, THR, 0, stream>>>(dinv, N);
    deg_count<<<(E + THR - 1) / THR, THR, 0, stream>>>(dstp, dinv, E);
    finalize_dinv<<<(N + THR - 1) / THR, THR, 0, stream>>>(dinv, N);

    const float* in = x;
    int K = NFEAT;
    for (int l = 0; l < 4; ++l) {
        const int Kpad = (K + 3) & ~3;
        const size_t shmem =
            ((size_t)(Kpad / 2) * WS2 * 2 + 8u * 32u * A_LD) * sizeof(float);
        // Fused GEMM + self-loop seed: r0 = in @ W ; r1 = r0*dinv^2 + b
        gcn_gemm_wmma_f32<<<(N + 255) / 256, 256, shmem, stream>>>(
            in, Wl[l], bl[l], dinv, r0, r1, N, K, Kpad);

        edge_scatter<<<(E + 7) / 8, 256, 0, stream>>>(srcp, dstp, dinv, r0, r1, E);

        float* outp = (l == 3) ? (float*)d_out : r2;
        tanh_act<<<((size_t)N * HDIM + THR - 1) / THR, THR, 0, stream>>>(
            r1, outp, (int)((size_t)N * HDIM));

        in = r2;
        K = HDIM;
    }
}